// SSMLayer_10411000725865
// MI455X (gfx1250) — compile-verified
//
#include <hip/hip_runtime.h>
#include <hip/hip_bf16.h>

typedef __attribute__((ext_vector_type(2))) float v2f;
typedef __attribute__((ext_vector_type(8))) float v8f;

#define HH 256
#define LL 4096
#define BBATCH 8
#define NST 64
#define QCH 64
#define NCH (LL / QCH)

// workspace layout (float offsets)
static const size_t F_UT = 0;                                  // (H,L,8)  transposed input
static const size_t F_YT = F_UT + (size_t)HH * LL * BBATCH;    // (H,L,8)  transposed output
static const size_t F_AB = F_YT + (size_t)HH * LL * BBATCH;    // (H,64,64) discretized A
static const size_t F_BB = F_AB + (size_t)HH * NST * NST;      // (H,64)    discretized B
static const size_t F_O  = F_BB + (size_t)HH * NST;            // (H,64,64) O[i,:] = C Ab^i
static const size_t F_W  = F_O  + (size_t)HH * NST * NST;      // (H,64,64) W[:,r] = Ab^{64-r} Bb
static const size_t F_P  = F_W  + (size_t)HH * NST * NST;      // (H,64,64) P = Ab^64
static const size_t F_K  = F_P  + (size_t)HH * NST * NST;      // (H,64)    K[d] = C Ab^d Bb

// ---------------------------------------------------------------------------
// Tensor Data Mover staging for the per-chunk U slab (512 floats, contiguous).
// D# packing per CDNA5 ISA.
// ---------------------------------------------------------------------------
#if __has_builtin(__builtin_amdgcn_tensor_load_to_lds) && __has_builtin(__builtin_amdgcn_s_wait_tensorcnt)
#define USE_TDM 1
typedef __attribute__((ext_vector_type(4))) unsigned int tdm_v4u;
typedef __attribute__((ext_vector_type(8))) int tdm_v8i;
typedef __attribute__((ext_vector_type(4))) int tdm_v4i;

__device__ inline void tdm_load_chunk(const float* src_global, void* dst_lds) {
  unsigned long long ga = (unsigned long long)(size_t)src_global;
  unsigned int lds = (unsigned int)(size_t)dst_lds;  // low 32 bits of generic ptr = LDS offset
  tdm_v4u g0;
  g0.x = 1u;                                              // count=1, user mode, no gather
  g0.y = lds;                                             // lds_addr
  g0.z = (unsigned int)ga;                                // global_addr[31:0]
  g0.w = (unsigned int)((ga >> 32) & 0x01FFFFFFull)       // global_addr[56:32]
         | 0x80000000u;                                   // type=2 ("image") at bits 127:126
  tdm_v8i g1;
  g1[0] = 0x00020000;            // workgroup_mask=0 (not in cluster), data_size=2 -> 4 bytes
  g1[1] = (int)(512u << 16);     // atomic_barrier_addr=0 | tensor_dim0[15:0]=512
  g1[2] = (int)(1u << 16);       // tensor_dim0[31:16]=0 | tensor_dim1[15:0]=1
  g1[3] = (int)(512u << 16);     // tensor_dim1[31:16]=0 | tile_dim0=512 (1-D tile)
  g1[4] = 0;                     // tile_dim1=0 (unused), tile_dim2=0 (unused)
  g1[5] = 512;                   // tensor_dim0_stride[31:0]
  g1[6] = 0;                     // tensor_dim0_stride[47:32] | tensor_dim1_stride[15:0]
  g1[7] = 0;                     // tensor_dim1_stride[47:16]
  tdm_v4i z4 = {0, 0, 0, 0};     // groups 2/3 unused (<=2D tensor)
#if __has_include(<hip/amd_detail/amd_gfx1250_TDM.h>)
  tdm_v8i z8 = {0, 0, 0, 0, 0, 0, 0, 0};
  __builtin_amdgcn_tensor_load_to_lds(g0, g1, z4, z4, z8, 0);  // clang-23 / therock arity
#else
  __builtin_amdgcn_tensor_load_to_lds(g0, g1, z4, z4, 0);      // ROCm 7.2 arity
#endif
}
#else
#define USE_TDM 0
#endif

// ---------------------------------------------------------------------------
// k0: transpose u (B,L,H) -> ut (H,L,8). Coalesced reads; L2 absorbs scatter.
// ---------------------------------------------------------------------------
__global__ __launch_bounds__(256) void k_transpose_in(const float* __restrict__ u,
                                                      float* __restrict__ ut) {
  size_t idx = (size_t)blockIdx.x * 256 + threadIdx.x;
  int h = (int)(idx & 255);
  int t = (int)((idx >> 8) & 4095);
  int b = (int)(idx >> 20);
  ut[((size_t)h << 15) + (size_t)t * 8 + b] = u[idx];
}

// ---------------------------------------------------------------------------
// k1: per-head bilinear discretization via Gauss-Jordan on [I-sA/2 | I+sA/2 | sB]
// ---------------------------------------------------------------------------
__global__ __launch_bounds__(256) void k_discretize(const float* __restrict__ A,
                                                    const float* __restrict__ Bv,
                                                    const float* __restrict__ logstep,
                                                    float* __restrict__ AbG,
                                                    float* __restrict__ BbG) {
  __shared__ float M[64][132];  // [ M | RHS(64) | b ] = 129 cols, padded
  int h = blockIdx.x, tid = threadIdx.x;
  float step = __expf(logstep[h]);
  const float* Ah = A + (size_t)h * 4096;
  for (int idx = tid; idx < 4096; idx += 256) {
    int i = idx >> 6, j = idx & 63;
    float a = 0.5f * step * Ah[idx];
    float d = (i == j) ? 1.0f : 0.0f;
    M[i][j]      = d - a;
    M[i][64 + j] = d + a;
  }
  if (tid < 64) M[tid][128] = step * Bv[h * 64 + tid];
  __syncthreads();

  for (int k = 0; k < 64; ++k) {
    float pinv = 1.0f / M[k][k];
    __syncthreads();
    for (int j = tid; j < 129; j += 256) M[k][j] *= pinv;
    __syncthreads();
    int i  = tid & 63;
    int c0 = (tid >> 6) * 33;
    int c1 = c0 + 33; if (c1 > 129) c1 = 129;
    float f = M[i][k];
    __syncthreads();
    if (i != k) {
      for (int j = c0; j < c1; ++j) M[i][j] -= f * M[k][j];
    }
    __syncthreads();
  }
  for (int idx = tid; idx < 4096; idx += 256) {
    int i = idx >> 6, j = idx & 63;
    AbG[(size_t)h * 4096 + idx] = M[i][64 + j];
  }
  if (tid < 64) BbG[h * 64 + tid] = M[tid][128];
}

// ---------------------------------------------------------------------------
// k2: per-head powers: O rows, W cols, K[0:64], P = Ab^64 (6 squarings)
// ---------------------------------------------------------------------------
__global__ __launch_bounds__(64) void k_powers(const float* __restrict__ AbG,
                                               const float* __restrict__ BbG,
                                               const float* __restrict__ Cin,
                                               float* __restrict__ Og,
                                               float* __restrict__ Wg,
                                               float* __restrict__ Pg,
                                               float* __restrict__ Kg) {
  __shared__ float Ab[64][65];
  __shared__ float M0[64][65], M1[64][65];
  __shared__ float v[64], ov[64], red[64];
  int h = blockIdx.x, t = threadIdx.x;
  for (int idx = t; idx < 4096; idx += 64) {
    int i = idx >> 6, j = idx & 63;
    float a = AbG[(size_t)h * 4096 + idx];
    Ab[i][j] = a; M0[i][j] = a;
  }
  float c = Cin[h * 64 + t];
  v[t]  = BbG[h * 64 + t];   // v_0 = Bb
  ov[t] = c;                 // o_0 = C
  __syncthreads();

  for (int i = 0; i < 64; ++i) {
    Og[(size_t)h * 4096 + (size_t)i * 64 + t] = ov[t];  // O[i,:] = C Ab^i
    red[t] = c * v[t];
    __syncthreads();
    if (t == 0) {
      float s = 0.0f;
      for (int j = 0; j < 64; ++j) s += red[j];
      Kg[h * 64 + i] = s;                               // K[i] = C Ab^i Bb
    }
    float nv = 0.0f, no = 0.0f;
    for (int j = 0; j < 64; ++j) {
      nv += Ab[t][j] * v[j];       // v_{i+1} = Ab v_i
      no += ov[j] * Ab[j][t];      // o_{i+1} = o_i Ab
    }
    __syncthreads();
    v[t] = nv; ov[t] = no;
    __syncthreads();
    Wg[(size_t)h * 4096 + (size_t)t * 64 + (63 - i)] = nv;  // W[:,63-i] = Ab^{64-(63-i)} Bb
  }

  // P = Ab^64 via six squarings (M0 -> M1 -> M0 -> ... -> M0)
  for (int s = 0; s < 6; ++s) {
    float (*src)[65] = (s & 1) ? M1 : M0;
    float (*dst)[65] = (s & 1) ? M0 : M1;
    float row[64];
    for (int j = 0; j < 64; ++j) row[j] = src[t][j];
    for (int j = 0; j < 64; ++j) {
      float acc = 0.0f;
      for (int m = 0; m < 64; ++m) acc += row[m] * src[m][j];
      dst[t][j] = acc;
    }
    __syncthreads();
  }
  for (int idx = t; idx < 4096; idx += 64) {
    int i = idx >> 6, j = idx & 63;
    Pg[(size_t)h * 4096 + idx] = M0[i][j];
  }
}

// ---------------------------------------------------------------------------
// k3: chunked scan with f32 WMMA 16x16x4.
//   per chunk:  Y = T64 @ U + O @ X   (waves 0-3)
//               X' = P @ X + W @ U    (waves 4-7)
// U staged by TDM (TENSORcnt) as a straight [t][b] slab; X lives transposed
// [n][state]; A-operands row-major stride 65 (bank-conflict free); Toeplitz
// operand generated on the fly from K[0:64].
// ---------------------------------------------------------------------------
__global__ __launch_bounds__(256) void k_scan(const float* __restrict__ ut,
                                              const float* __restrict__ Og,
                                              const float* __restrict__ Wg,
                                              const float* __restrict__ Pg,
                                              const float* __restrict__ Kg,
                                              float* __restrict__ yt) {
  __shared__ __align__(16) float Om[64 * 65];
  __shared__ __align__(16) float Wm[64 * 65];
  __shared__ __align__(16) float Pm[64 * 65];
  __shared__ __align__(16) float Ub[2][528];      // [time-in-chunk][batch], +16 pad for n>=8 overread
  __shared__ __align__(16) float Xb[16 * 65];     // [batch(padded 16)][state]
  __shared__ float Kl[65];
  int h = blockIdx.x, tid = threadIdx.x;
  int lane = tid & 31, wv = tid >> 5;

  for (int idx = tid; idx < 4096; idx += 256) {
    int i = idx >> 6, j = idx & 63;
    Om[i * 65 + j] = Og[(size_t)h * 4096 + idx];
    Wm[i * 65 + j] = Wg[(size_t)h * 4096 + idx];
    Pm[i * 65 + j] = Pg[(size_t)h * 4096 + idx];
  }
  if (tid < 64) Kl[tid] = Kg[h * 64 + tid];
  for (int idx = tid; idx < 16 * 65; idx += 256) Xb[idx] = 0.0f;
  __syncthreads();

  const float* uh = ut + (size_t)h * (LL * BBATCH);
  // preload chunk 0 (512 contiguous floats)
#if USE_TDM
  if (wv == 0) {
    tdm_load_chunk(uh, &Ub[0][0]);
    __builtin_amdgcn_s_wait_tensorcnt(0);
  }
#else
  {
    int i0 = tid * 2;
    *reinterpret_cast<float2*>(&Ub[0][i0]) = *reinterpret_cast<const float2*>(uh + i0);
  }
#endif
  __syncthreads();

  int mB = (wv & 3) * 16;
  int m  = mB + (lane & 15);       // A-fragment row
  int kh = (lane >> 4) * 2;        // K sub-pair: lanes 0-15 -> K0,K1; 16-31 -> K2,K3
  int n  = lane & 15;              // B/C-fragment column

  for (int j = 0; j < NCH; ++j) {
#if USE_TDM
    // fire-and-forget DMA of next chunk; latency hidden behind this chunk's WMMAs
    if (j + 1 < NCH && wv == 0) tdm_load_chunk(uh + (size_t)(j + 1) * QCH * BBATCH, &Ub[(j + 1) & 1][0]);
#endif
    const float* U = Ub[j & 1];
    v8f acc = {0.f, 0.f, 0.f, 0.f, 0.f, 0.f, 0.f, 0.f};

    if (wv < 4) {
      // Y tile: T64 @ U  (Toeplitz from Kl) + O @ X
      #pragma unroll
      for (int k = 0; k < 64; k += 4) {
        int k0 = k + kh;
        v2f a, b;
        a.x = (m >= k0)     ? Kl[m - k0]     : 0.0f;
        a.y = (m >= k0 + 1) ? Kl[m - k0 - 1] : 0.0f;
        b.x = U[k0 * 8 + n]; b.y = U[k0 * 8 + 8 + n];
        acc = __builtin_amdgcn_wmma_f32_16x16x4_f32(false, a, false, b, (short)0, acc, false, false);
      }
      #pragma unroll
      for (int k = 0; k < 64; k += 4) {
        int k0 = k + kh;
        v2f a, b;
        a.x = Om[m * 65 + k0]; a.y = Om[m * 65 + k0 + 1];
        b.x = Xb[n * 65 + k0]; b.y = Xb[n * 65 + k0 + 1];
        acc = __builtin_amdgcn_wmma_f32_16x16x4_f32(false, a, false, b, (short)0, acc, false, false);
      }
    } else {
      // X' tile: P @ X + W @ U
      #pragma unroll
      for (int k = 0; k < 64; k += 4) {
        int k0 = k + kh;
        v2f a, b;
        a.x = Pm[m * 65 + k0]; a.y = Pm[m * 65 + k0 + 1];
        b.x = Xb[n * 65 + k0]; b.y = Xb[n * 65 + k0 + 1];
        acc = __builtin_amdgcn_wmma_f32_16x16x4_f32(false, a, false, b, (short)0, acc, false, false);
      }
      #pragma unroll
      for (int k = 0; k < 64; k += 4) {
        int k0 = k + kh;
        v2f a, b;
        a.x = Wm[m * 65 + k0]; a.y = Wm[m * 65 + k0 + 1];
        b.x = U[k0 * 8 + n];   b.y = U[k0 * 8 + 8 + n];
        acc = __builtin_amdgcn_wmma_f32_16x16x4_f32(false, a, false, b, (short)0, acc, false, false);
      }
    }
    __syncthreads();  // all reads of Xb / Ub[j&1] complete

    if (wv < 4) {
      if (n < 8) {  // columns 8..15 are batch padding (column-independent, never stored)
        #pragma unroll
        for (int r = 0; r < 8; ++r) {
          int trow = mB + r + 8 * (lane >> 4);  // C/D layout: VGPR r -> M = r + 8*(lane/16)
          yt[(size_t)h * (LL * BBATCH) + (size_t)(j * QCH + trow) * BBATCH + n] = acc[r];
        }
      }
    } else {
      #pragma unroll
      for (int r = 0; r < 8; ++r) {
        int srow = mB + r + 8 * (lane >> 4);
        Xb[n * 65 + srow] = acc[r];  // store X' transposed for next chunk's B-operand
      }
    }

#if USE_TDM
    if (wv == 0) __builtin_amdgcn_s_wait_tensorcnt(0);  // next chunk's DMA landed
#else
    if (j + 1 < NCH) {  // double-buffered fallback copy of next chunk
      const float* src = uh + (size_t)(j + 1) * QCH * BBATCH;
      int i0 = tid * 2;
      float2 val = *reinterpret_cast<const float2*>(src + i0);
      float* Ud = Ub[(j + 1) & 1];
      Ud[i0]     = val.x;
      Ud[i0 + 1] = val.y;
      if (j + 2 < NCH) __builtin_prefetch(uh + (size_t)(j + 2) * QCH * BBATCH + i0, 0, 1);
    }
#endif
    __syncthreads();  // X' and next U visible to all waves
  }
}

// ---------------------------------------------------------------------------
// k4: transpose back + skip connection:  out[b,t,h] = yt[h,t,b] + D[h]*u[b,t,h]
// ---------------------------------------------------------------------------
__global__ __launch_bounds__(256) void k_out(const float* __restrict__ yt,
                                             const float* __restrict__ u,
                                             const float* __restrict__ D,
                                             float* __restrict__ out) {
  size_t idx = (size_t)blockIdx.x * 256 + threadIdx.x;
  int h = (int)(idx & 255);
  int t = (int)((idx >> 8) & 4095);
  int b = (int)(idx >> 20);
  out[idx] = yt[((size_t)h << 15) + (size_t)t * 8 + b] + D[h] * u[idx];
}

// ---------------------------------------------------------------------------
extern "C" void kernel_launch(void* const* d_in, const int* in_sizes, int n_in,
                              void* d_out, int out_size, void* d_ws, size_t ws_size,
                              hipStream_t stream) {
  (void)in_sizes; (void)n_in; (void)out_size; (void)ws_size;
  const float* u  = (const float*)d_in[0];
  const float* A  = (const float*)d_in[1];
  const float* Bv = (const float*)d_in[2];
  const float* Cv = (const float*)d_in[3];
  const float* Dv = (const float*)d_in[4];
  const float* ls = (const float*)d_in[5];
  float* out = (float*)d_out;

  float* ws = (float*)d_ws;
  float* ut = ws + F_UT;
  float* yt = ws + F_YT;
  float* Ab = ws + F_AB;
  float* Bb = ws + F_BB;
  float* O  = ws + F_O;
  float* W  = ws + F_W;
  float* P  = ws + F_P;
  float* K  = ws + F_K;

  int nElem = BBATCH * LL * HH;  // 8,388,608
  k_transpose_in<<<dim3(nElem / 256), dim3(256), 0, stream>>>(u, ut);
  k_discretize<<<dim3(HH), dim3(256), 0, stream>>>(A, Bv, ls, Ab, Bb);
  k_powers<<<dim3(HH), dim3(64), 0, stream>>>(Ab, Bb, Cv, O, W, P, K);
  k_scan<<<dim3(HH), dim3(256), 0, stream>>>(ut, O, W, P, K, yt);
  k_out<<<dim3(nElem / 256), dim3(256), 0, stream>>>(yt, u, Dv, out);
}